// CD_BP_Net_21663815041317
// MI455X (gfx1250) — compile-verified
//
#include <hip/hip_runtime.h>
#include <math.h>

typedef __attribute__((ext_vector_type(2))) float v2f;
typedef __attribute__((ext_vector_type(8))) float v8f;

#define TPB 256

// ---------------- init: zero scalars (ws[0..15]) and deg[N] ----------------
__global__ void k_init(float* scal, float* deg, int N) {
    int i = blockIdx.x * blockDim.x + threadIdx.x;
    if (i < 16) scal[i] = 0.f;
    if (i < N)  deg[i]  = 0.f;
}

// ---------------- prep: two_m, deg, expm1(beta*w), psi0 = msg/rowsum -------
__global__ void k_prep(const long long* __restrict__ eidx,
                       const float* __restrict__ attr,
                       const float* __restrict__ msg,
                       const float* __restrict__ beta_p,
                       float* __restrict__ deg,
                       float* __restrict__ two_m,
                       float* __restrict__ expm1w,
                       float* __restrict__ psi,
                       int E) {
    int e = blockIdx.x * blockDim.x + threadIdx.x;
    float w = 0.f;
    if (e < E) {
        w = attr[e];
        int src = (int)eidx[e];
        atomicAdd(&deg[src], w);
        float beta = beta_p[0];
        expm1w[e] = expm1f(beta * w);
        float4 m = ((const float4*)msg)[e];
        float inv = 1.f / (m.x + m.y + m.z + m.w);
        ((float4*)psi)[e] = make_float4(m.x * inv, m.y * inv, m.z * inv, m.w * inv);
    }
    // block reduction of w -> one atomic per block for two_m
    __shared__ float red[TPB / 32];
    float t = w;
    #pragma unroll
    for (int off = 16; off; off >>= 1) t += __shfl_xor(t, off, 32);
    int lane = threadIdx.x & 31, wid = threadIdx.x >> 5;
    if (lane == 0) red[wid] = t;
    __syncthreads();
    if (threadIdx.x == 0) {
        float s = 0.f;
        #pragma unroll
        for (int i = 0; i < TPB / 32; ++i) s += red[i];
        atomicAdd(two_m, s);
    }
}

// ---------------- zero a float array ---------------------------------------
__global__ void k_zero(float* __restrict__ p, int n) {
    int i = blockIdx.x * blockDim.x + threadIdx.x;
    if (i < n) p[i] = 0.f;
}

// ---------------- edge pass 1: L = log1p(psi*ew); scatter-add into S -------
__global__ void k_edge_L(const long long* __restrict__ eidx,
                         const float* __restrict__ expm1w,
                         const float* __restrict__ psi,
                         float* __restrict__ L,
                         float* __restrict__ S,
                         int E) {
    int e = blockIdx.x * blockDim.x + threadIdx.x;
    if (e >= E) return;
    int pf = e + 8192;                       // stream-ahead prefetch (gfx1250 global_prefetch_b8)
    if (pf < E) __builtin_prefetch(psi + 4 * (size_t)pf, 0, 0);
    float4 p = ((const float4*)psi)[e];
    float ew = expm1w[e];
    float4 l = make_float4(log1pf(p.x * ew), log1pf(p.y * ew),
                           log1pf(p.z * ew), log1pf(p.w * ew));
    ((float4*)L)[e] = l;
    int dst = (int)eidx[E + e];
    float* sp = S + 4 * (size_t)dst;         // S is 1.6MB -> L2-resident atomics
    atomicAdd(sp + 0, l.x);
    atomicAdd(sp + 1, l.y);
    atomicAdd(sp + 2, l.z);
    atomicAdd(sp + 3, l.w);
}

// ---------------- node pass: marg = softmax(h+S); hacc += deg*marg (WMMA) --
__global__ void k_node(const float* __restrict__ S,
                       const float* __restrict__ deg,
                       const float* __restrict__ h,
                       float* __restrict__ hacc,
                       float* __restrict__ out_s,
                       int N, int write_out) {
    int n = blockIdx.x * blockDim.x + threadIdx.x;
    bool valid = (n < N);
    float4 Sn = valid ? ((const float4*)S)[n] : make_float4(0.f, 0.f, 0.f, 0.f);
    float l0 = h[0] + Sn.x, l1 = h[1] + Sn.y, l2 = h[2] + Sn.z, l3 = h[3] + Sn.w;
    float mx = fmaxf(fmaxf(l0, l1), fmaxf(l2, l3));
    float e0 = expf(l0 - mx), e1 = expf(l1 - mx), e2 = expf(l2 - mx), e3 = expf(l3 - mx);
    float inv = 1.f / (e0 + e1 + e2 + e3);
    float m0 = e0 * inv, m1 = e1 * inv, m2 = e2 * inv, m3 = e3 * inv;
    if (valid && write_out) ((float4*)out_s)[n] = make_float4(m0, m1, m2, m3);
    float dg = valid ? deg[n] : 0.f;
    float v0 = dg * m0, v1 = dg * m1, v2 = dg * m2, v3 = dg * m3;

    // --- exact f32 wave reduction via V_WMMA_F32_16X16X4_F32 ---
    // fold 32->16 lanes; lane l then holds fold_c[l&15] for each community c
    v0 += __shfl_xor(v0, 16, 32);
    v1 += __shfl_xor(v1, 16, 32);
    v2 += __shfl_xor(v2, 16, 32);
    v3 += __shfl_xor(v3, 16, 32);
    int lane = threadIdx.x & 31;
    // A (16x4 f32 layout): VGPR0 = {K=0 | K=2}, VGPR1 = {K=1 | K=3}; M = lane&15
    v2f a;
    a.x = (lane < 16) ? v0 : v2;
    a.y = (lane < 16) ? v1 : v3;
    // B = 4x4 identity embedded in 4x16 layout: VGPR0 = {K=0 row | K=2 row}, VGPR1 = {K=1 | K=3}
    v2f b;
    b.x = (lane == 0 || lane == 18) ? 1.f : 0.f;
    b.y = (lane == 1 || lane == 19) ? 1.f : 0.f;
    v8f c = {};
    // D[m][n] = sum_k A[m][k]*I[k][n] = fold_n[m]  (EXEC is all-ones here)
    c = __builtin_amdgcn_wmma_f32_16x16x4_f32(false, a, false, b, (short)0, c, false, false);
    float rp = c[0] + c[1] + c[2] + c[3] + c[4] + c[5] + c[6] + c[7];
    rp += __shfl_xor(rp, 16, 32);            // add M=8..15 half (lanes 16..31)
    if (lane < 4) atomicAdd(&hacc[lane], rp);  // lane c holds sum_n deg[n]*marg[n][c]
}

// ---------------- tiny: h = -(beta/two_m) * hacc; reset hacc ---------------
__global__ void k_hupd(const float* __restrict__ beta_p,
                       const float* __restrict__ two_m,
                       float* __restrict__ hacc,
                       float* __restrict__ h) {
    int k = threadIdx.x;
    if (k < 4) {
        float coef = -beta_p[0] / two_m[0];
        h[k] = coef * hacc[k];
        hacc[k] = 0.f;
    }
}

// ---------------- edge pass 2: psi = softmax(h + S[src] - L[rev]) ----------
__global__ void k_edge_psi(const long long* __restrict__ eidx,
                           const float* __restrict__ S,
                           const float* __restrict__ L,
                           const float* __restrict__ h,
                           float* __restrict__ psi,
                           int E, int e_und) {
    int e = blockIdx.x * blockDim.x + threadIdx.x;
    if (e >= E) return;
    int rev = (e < e_und) ? (e + e_und) : (e - e_und);   // streaming +-E/2 offset
    int pf = e + 8192;
    if (pf < E) {
        int pr = (pf < e_und) ? (pf + e_und) : (pf - e_und);
        __builtin_prefetch(L + 4 * (size_t)pr, 0, 0);
    }
    float4 Lr = ((const float4*)L)[rev];
    int src = (int)eidx[e];
    float4 Ss = ((const float4*)S)[src];                  // L2-resident gather
    float l0 = h[0] + Ss.x - Lr.x;
    float l1 = h[1] + Ss.y - Lr.y;
    float l2 = h[2] + Ss.z - Lr.z;
    float l3 = h[3] + Ss.w - Lr.w;
    float mx = fmaxf(fmaxf(l0, l1), fmaxf(l2, l3));
    float e0 = expf(l0 - mx), e1 = expf(l1 - mx), e2 = expf(l2 - mx), e3 = expf(l3 - mx);
    float inv = 1.f / (e0 + e1 + e2 + e3);
    ((float4*)psi)[e] = make_float4(e0 * inv, e1 * inv, e2 * inv, e3 * inv);
}

// ---------------- modularity link term -------------------------------------
__global__ void k_link(const long long* __restrict__ eidx,
                       const float* __restrict__ attr,
                       const float* __restrict__ s,
                       float* __restrict__ link,
                       int E) {
    int e = blockIdx.x * blockDim.x + threadIdx.x;
    float part = 0.f;
    if (e < E) {
        int a = (int)eidx[e], b = (int)eidx[E + e];
        float4 sa = ((const float4*)s)[a];
        float4 sb = ((const float4*)s)[b];
        part = attr[e] * (sa.x * sb.x + sa.y * sb.y + sa.z * sb.z + sa.w * sb.w);
    }
    __shared__ float red[TPB / 32];
    #pragma unroll
    for (int off = 16; off; off >>= 1) part += __shfl_xor(part, off, 32);
    int lane = threadIdx.x & 31, wid = threadIdx.x >> 5;
    if (lane == 0) red[wid] = part;
    __syncthreads();
    if (threadIdx.x == 0) {
        float t = 0.f;
        #pragma unroll
        for (int i = 0; i < TPB / 32; ++i) t += red[i];
        atomicAdd(link, t);
    }
}

// ---------------- q_mod = link/2m - sum((deg·s/2m)^2) ----------------------
__global__ void k_final(const float* __restrict__ two_m,
                        const float* __restrict__ link,
                        const float* __restrict__ hacc,
                        float* __restrict__ out_q) {
    if (threadIdx.x == 0) {
        float tm = two_m[0];
        float c0 = hacc[0] / tm, c1 = hacc[1] / tm, c2 = hacc[2] / tm, c3 = hacc[3] / tm;
        out_q[0] = link[0] / tm - (c0 * c0 + c1 * c1 + c2 * c2 + c3 * c3);
    }
}

extern "C" void kernel_launch(void* const* d_in, const int* in_sizes, int n_in,
                              void* d_out, int out_size, void* d_ws, size_t ws_size,
                              hipStream_t stream) {
    const long long* eidx  = (const long long*)d_in[0];   // [2,E] int64
    const float*     attr  = (const float*)d_in[1];       // [E]
    const float*     msg   = (const float*)d_in[2];       // [E,4]
    const float*     betap = (const float*)d_in[3];       // [1]

    const int E = in_sizes[1];
    const int N = (out_size - 1) / 4;                     // out = s[N,4] ++ q_mod
    const int e_und = E / 2;

    float* ws     = (float*)d_ws;
    float* two_m  = ws + 0;
    float* hacc   = ws + 1;     // 4
    float* hfld   = ws + 5;     // 4
    float* link   = ws + 9;     // 1
    float* deg    = ws + 16;                       // N      (64B-aligned)
    float* S      = deg + (size_t)N;               // 4N
    float* expm1w = S + 4 * (size_t)N;             // E
    float* Lbuf   = expm1w + (size_t)E;            // 4E
    float* psi    = Lbuf + 4 * (size_t)E;          // 4E

    float* out_s = (float*)d_out;
    float* out_q = out_s + 4 * (size_t)N;

    dim3 blk(TPB);
    int gE = (E + TPB - 1) / TPB;
    int gN = (N + TPB - 1) / TPB;
    int gZ = (4 * N + TPB - 1) / TPB;
    int gI = ((N > 16 ? N : 16) + TPB - 1) / TPB;

    k_init<<<gI, blk, 0, stream>>>(ws, deg, N);
    k_prep<<<gE, blk, 0, stream>>>(eidx, attr, msg, betap, deg, two_m, expm1w, psi, E);

    for (int it = 0; it < 10; ++it) {
        k_zero<<<gZ, blk, 0, stream>>>(S, 4 * N);
        k_edge_L<<<gE, blk, 0, stream>>>(eidx, expm1w, psi, Lbuf, S, E);
        k_node<<<gN, blk, 0, stream>>>(S, deg, hfld, hacc, out_s, N, 0);
        k_hupd<<<1, 32, 0, stream>>>(betap, two_m, hacc, hfld);
        k_edge_psi<<<gE, blk, 0, stream>>>(eidx, S, Lbuf, hfld, psi, E, e_und);
    }

    // final marginals + modularity
    k_zero<<<gZ, blk, 0, stream>>>(S, 4 * N);
    k_edge_L<<<gE, blk, 0, stream>>>(eidx, expm1w, psi, Lbuf, S, E);
    k_node<<<gN, blk, 0, stream>>>(S, deg, hfld, hacc, out_s, N, 1);  // hacc = sum deg*s
    k_link<<<gE, blk, 0, stream>>>(eidx, attr, out_s, link, E);
    k_final<<<1, 32, 0, stream>>>(two_m, link, hacc, out_q);
}